// _QuantumFeedForward_65481071396273
// MI455X (gfx1250) — compile-verified
//
#include <hip/hip_runtime.h>

typedef float v2f __attribute__((ext_vector_type(2)));
typedef float v8f __attribute__((ext_vector_type(8)));

#define EMBED 1024
#define NQ 8
#define WAVES_PER_WG 8
#define TOK_PER_WAVE 16
#define TOK_PER_WG (WAVES_PER_WG * TOK_PER_WAVE)

__device__ __forceinline__ float fcos(float a) {
  // v_cos_f32 takes revolutions; xang/theta are small so no range reduction needed
  return __builtin_amdgcn_cosf(a * 0.15915494309189535f);
}

__global__ __launch_bounds__(256) void qff_kernel(
    const float* __restrict__ x, const float* __restrict__ W_in,
    const float* __restrict__ b_in, const float* __restrict__ theta,
    const float* __restrict__ W_out, const float* __restrict__ b_out,
    float* __restrict__ out)
{
  const int tid  = threadIdx.x;
  const int lane = tid & 31;
  const int wave = tid >> 5;
  const int l16  = lane & 15;
  const int hi   = lane >> 4;                 // 0 = lanes 0-15, 1 = lanes 16-31
  const int tokb = (blockIdx.x * WAVES_PER_WG + wave) * TOK_PER_WAVE;

  // ---------------- GEMM1: C[q][t] = sum_k W_in[q][k] * x[t][k]  (+ b_in) ----------------
  // A (16x4 f32): lane m = l16, rows 8..15 duplicate rows 0..7 so ALL lanes end up
  //               holding the full xang vector of their token.
  // B (4x16 f32): lane n = l16 = token; VGPR0 holds K=0(lo)/K=2(hi), VGPR1 K=1/K=3.
  const float* xrow = x    + (size_t)(tokb + l16) * EMBED + 2 * hi;
  const float* wrow = W_in + (size_t)(l16 & 7)    * EMBED + 2 * hi;

  v8f c0 = {}, c1 = {};
  #pragma unroll
  for (int r = 0; r < NQ; ++r) c0[r] = b_in[r];   // bias folded into accumulator

  #pragma unroll 4
  for (int kt = 0; kt < 256; kt += 2) {           // 256 WMMAs of K=4 -> K=1024, 2 chains
    v2f a0 = *(const v2f*)(wrow + kt * 4);
    v2f b0 = *(const v2f*)(xrow + kt * 4);
    v2f a1 = *(const v2f*)(wrow + kt * 4 + 4);
    v2f b1 = *(const v2f*)(xrow + kt * 4 + 4);
    if ((kt & 15) == 0) __builtin_prefetch(xrow + kt * 4 + 512, 0, 1);
    c0 = __builtin_amdgcn_wmma_f32_16x16x4_f32(false, a0, false, b0, (short)0, c0, false, false);
    c1 = __builtin_amdgcn_wmma_f32_16x16x4_f32(false, a1, false, b1, (short)0, c1, false, false);
  }
  v8f cc = c0 + c1;   // merge the two accumulation chains

  // ---------------- quantum layer: z_w = prod_{j<=w} cos(theta_j)*cos(xang_j) ------------
  float z[NQ];
  {
    float p = 1.0f;
    #pragma unroll
    for (int r = 0; r < NQ; ++r) {
      p *= fcos(theta[r]) * fcos(cc[r]);
      z[r] = p;
    }
  }

  // A operands for GEMM2 (z is A, 16x8 -> two 16x4 slabs)
  v2f A1 = { hi ? z[2] : z[0], hi ? z[3] : z[1] };
  v2f A2 = { hi ? z[6] : z[4], hi ? z[7] : z[5] };

  // ---------------- GEMM2: out[t][e] = sum_q z[t][q] * W_out[e][q] + b_out[e], ReLU ------
  // B (4x16): lane n = l16 = output column within tile; values W_out[n0+n][k], k by half.
  const float* wo    = W_out + (size_t)l16 * NQ + 2 * hi;   // + n0t*16*NQ per tile
  float*       obase = out + (size_t)(tokb + 8 * hi) * EMBED + l16;

  for (int n0t = 0; n0t < EMBED / 16; ++n0t) {
    v2f B1 = *(const v2f*)(wo + (size_t)n0t * 16 * NQ);
    v2f B2 = *(const v2f*)(wo + (size_t)n0t * 16 * NQ + 4);
    float bo = b_out[n0t * 16 + l16];
    v8f d;
    #pragma unroll
    for (int r = 0; r < 8; ++r) d[r] = bo;        // bias folded into accumulator
    d = __builtin_amdgcn_wmma_f32_16x16x4_f32(false, A1, false, B1, (short)0, d, false, false);
    d = __builtin_amdgcn_wmma_f32_16x16x4_f32(false, A2, false, B2, (short)0, d, false, false);
    #pragma unroll
    for (int r = 0; r < 8; ++r) {                 // C: VGPR r holds row r (lo) / r+8 (hi)
      obase[(size_t)r * EMBED + n0t * 16] = fmaxf(d[r], 0.0f);
    }
  }
}

extern "C" void kernel_launch(void* const* d_in, const int* in_sizes, int n_in,
                              void* d_out, int out_size, void* d_ws, size_t ws_size,
                              hipStream_t stream) {
  const float* x     = (const float*)d_in[0];
  const float* W_in  = (const float*)d_in[1];
  const float* b_in  = (const float*)d_in[2];
  const float* theta = (const float*)d_in[3];
  const float* W_out = (const float*)d_in[4];
  const float* b_out = (const float*)d_in[5];
  float* out = (float*)d_out;

  const int n_tok = in_sizes[0] / EMBED;          // 32768
  const int grid  = n_tok / TOK_PER_WG;           // 256 workgroups, no tail
  qff_kernel<<<grid, 256, 0, stream>>>(x, W_in, b_in, theta, W_out, b_out, out);
}